// RelationalHypergraphLayer_37177236914497
// MI455X (gfx1250) — compile-verified
//
#include <hip/hip_runtime.h>
#include <hip/hip_bf16.h>
#include <math.h>

typedef __attribute__((ext_vector_type(2))) float v2f;
typedef __attribute__((ext_vector_type(8))) float v8f;

#define DMODEL 128
#define NHEAD  8
#define DHEAD  16

// ---------------------------------------------------------------------------
// utility: CAS-based float atomic max (scores may be negative)
// ---------------------------------------------------------------------------
__device__ __forceinline__ void atomicMaxF(float* addr, float val) {
  unsigned int* ua = (unsigned int*)addr;
  unsigned int old = __float_as_uint(*addr);
  while (__uint_as_float(old) < val) {
    unsigned int assumed = old;
    old = atomicCAS(ua, assumed, __float_as_uint(val));
    if (old == assumed) break;
  }
}

// ---------------------------------------------------------------------------
// fill
// ---------------------------------------------------------------------------
__global__ void fill_f32(float* __restrict__ p, float v, long n) {
  long i = (long)blockIdx.x * blockDim.x + threadIdx.x;
  if (i < n) p[i] = v;
}

// ---------------------------------------------------------------------------
// WMMA f32 GEMM:  Y[M x Nout] = X[M x Kdim] @ W[Kdim x Nout] + bias (+ReLU)
//
// Block = 256 threads (8 wave32s). Block tile = 128 rows x 16 cols; wave w
// owns rows [blockIdx.x*128 + w*16, +16). The 16-column W panel (Kdim x 16)
// is staged once per block into LDS via CDNA5 async global->LDS loads
// (GLOBAL_LOAD_ASYNC_TO_LDS_B32, per-lane LDS dest address -> relayout for
// free), drained with s_wait_asynccnt, then consumed as ds_load_b64 pairs.
//
// LDS layout: pair-interleaved: wlds[(k>>1)*32 + 2*c + (k&1)] = W[k][col0+c]
// so the B fragment for lane (half,r) at step k0 is the b64 at element
// (k0/2 + half)*32 + 2*r  (8-byte aligned, immediate-offset ds_load_b64).
//
// WMMA 16x16x4 f32 fragment mapping (wave32):
//   A (16x4): lane r=l&15 -> row r, half=l>>4 selects K pair {2h, 2h+1}
//   B (4x16): vgpr0 = K row 2h (cols on lanes), vgpr1 = K row 2h+1
//   D (16x16): vgpr v, lanes 0-15 -> M=v, lanes 16-31 -> M=v+8
// ---------------------------------------------------------------------------
__global__ void gemm_bias_wmma(const float* __restrict__ X,
                               const float* __restrict__ W,
                               const float* __restrict__ B,
                               float* __restrict__ Y,
                               int M, int Kdim, int Nout, int relu) {
  extern __shared__ float wlds[];
  const int col0 = blockIdx.y * 16;

  // ---- async stage of the W panel into LDS (pair-interleaved) ----
  for (int idx = threadIdx.x; idx < Kdim * 16; idx += 256) {
    const int k = idx >> 4;
    const int c = idx & 15;
    const float* gp = W + (size_t)k * Nout + col0 + c;
    unsigned ldsoff = (unsigned)(uintptr_t)&wlds[(k >> 1) * 32 + 2 * c + (k & 1)];
    asm volatile("global_load_async_to_lds_b32 %0, %1, off"
                 :: "v"(ldsoff), "v"(gp) : "memory");
  }
  asm volatile("s_wait_asynccnt 0" ::: "memory");
  __syncthreads();

  const int lid  = threadIdx.x & 31;
  const int wv   = threadIdx.x >> 5;                 // 0..7
  const int row0 = (blockIdx.x * 8 + wv) * 16;
  if (row0 >= M) return;                             // tail row-tiles (uniform per wave)
  const int half = lid >> 4;                         // 0/1
  const int r    = lid & 15;

  const float* Xrow = X + (size_t)(row0 + r) * Kdim + 2 * half;
  const float* Wp   = wlds + half * 32 + 2 * r;

  v8f acc = {};
  for (int k0 = 0; k0 < Kdim; k0 += 4) {
    v2f a = *(const v2f*)(Xrow + k0);                // global_load_b64, imm offsets
    v2f b = *(const v2f*)(Wp + (k0 >> 1) * 32);      // ds_load_b64, imm offsets
    acc = __builtin_amdgcn_wmma_f32_16x16x4_f32(false, a, false, b,
                                                (short)0, acc, false, false);
  }

  const float bias = B[col0 + r];
#pragma unroll
  for (int v = 0; v < 8; ++v) {
    float val = acc[v] + bias;
    if (relu) val = fmaxf(val, 0.0f);
    Y[(size_t)(row0 + half * 8 + v) * Nout + col0 + r] = val;
  }
}

// ---------------------------------------------------------------------------
// edge score: score[e,h] = dot(Q[src[e],h,:], K[dst[e],h,:]) * 0.25
// + atomic segment max into smax[dst,h].  One thread per (e,h).
// ---------------------------------------------------------------------------
__global__ void edge_score_kernel(const float* __restrict__ Q,
                                  const float* __restrict__ Km,
                                  const int* __restrict__ src,
                                  const int* __restrict__ dst,
                                  float* __restrict__ score,
                                  float* __restrict__ smax, int E) {
  int idx = blockIdx.x * blockDim.x + threadIdx.x;
  if (idx >= E * NHEAD) return;
  int e = idx >> 3, hh = idx & 7;
  int s = src[e], d = dst[e];
  const float4* q = (const float4*)(Q + (size_t)s * DMODEL + hh * DHEAD);
  const float4* k = (const float4*)(Km + (size_t)d * DMODEL + hh * DHEAD);
  float acc = 0.0f;
#pragma unroll
  for (int i = 0; i < 4; ++i) {
    float4 a = q[i], b = k[i];
    acc += a.x * b.x + a.y * b.y + a.z * b.z + a.w * b.w;
  }
  acc *= 0.25f;                                  // DH^-0.5 = 16^-0.5
  score[idx] = acc;
  atomicMaxF(smax + (size_t)d * NHEAD + hh, acc);
}

// ---------------------------------------------------------------------------
// ex = exp(score - smax[dst]);  denom[dst] += ex  (score overwritten in place)
// ---------------------------------------------------------------------------
__global__ void edge_exp_kernel(const int* __restrict__ dst,
                                float* __restrict__ score,
                                const float* __restrict__ smax,
                                float* __restrict__ denom, int E) {
  int idx = blockIdx.x * blockDim.x + threadIdx.x;
  if (idx >= E * NHEAD) return;
  int e = idx >> 3, hh = idx & 7;
  int d = dst[e];
  float ex = __expf(score[idx] - smax[(size_t)d * NHEAD + hh]);
  score[idx] = ex;
  atomicAdd(denom + (size_t)d * NHEAD + hh, ex);
}

// ---------------------------------------------------------------------------
// weighted scatter of V[src] into agg[dst] with a = ex/denom[dst]
// ---------------------------------------------------------------------------
__global__ void edge_scatter_kernel(const float* __restrict__ V,
                                    const int* __restrict__ src,
                                    const int* __restrict__ dst,
                                    const float* __restrict__ score,
                                    const float* __restrict__ denom,
                                    float* __restrict__ agg, int E) {
  int idx = blockIdx.x * blockDim.x + threadIdx.x;
  if (idx >= E * NHEAD) return;
  int e = idx >> 3, hh = idx & 7;
  int s = src[e], d = dst[e];
  float a = score[idx] / denom[(size_t)d * NHEAD + hh];
  const float4* v = (const float4*)(V + (size_t)s * DMODEL + hh * DHEAD);
  float* o = agg + (size_t)d * DMODEL + hh * DHEAD;
#pragma unroll
  for (int i = 0; i < 4; ++i) {
    float4 vv = v[i];
    atomicAdd(o + 4 * i + 0, vv.x * a);
    atomicAdd(o + 4 * i + 1, vv.y * a);
    atomicAdd(o + 4 * i + 2, vv.z * a);
    atomicAdd(o + 4 * i + 3, vv.w * a);
  }
}

// ---------------------------------------------------------------------------
// out = LN(X + R) * G + Bt  -- one wave32 per 128-wide row, 4 floats/lane
// ---------------------------------------------------------------------------
__global__ void add_ln_kernel(const float* __restrict__ X,
                              const float* __restrict__ R,
                              const float* __restrict__ G,
                              const float* __restrict__ Bt,
                              float* __restrict__ O, int nrows) {
  int wave = blockIdx.x * (blockDim.x >> 5) + (threadIdx.x >> 5);
  int lane = threadIdx.x & 31;
  if (wave >= nrows) return;
  float4 xv = ((const float4*)(X + (size_t)wave * DMODEL))[lane];
  float4 rv = ((const float4*)(R + (size_t)wave * DMODEL))[lane];
  float4 s = make_float4(xv.x + rv.x, xv.y + rv.y, xv.z + rv.z, xv.w + rv.w);
  float sum = s.x + s.y + s.z + s.w;
  float sq  = s.x * s.x + s.y * s.y + s.z * s.z + s.w * s.w;
#pragma unroll
  for (int off = 16; off > 0; off >>= 1) {
    sum += __shfl_xor(sum, off, 32);
    sq  += __shfl_xor(sq,  off, 32);
  }
  float mean = sum * (1.0f / DMODEL);
  float var  = sq * (1.0f / DMODEL) - mean * mean;
  float rstd = rsqrtf(var + 1e-5f);
  float4 gv = ((const float4*)G)[lane];
  float4 bv = ((const float4*)Bt)[lane];
  float4 o;
  o.x = (s.x - mean) * rstd * gv.x + bv.x;
  o.y = (s.y - mean) * rstd * gv.y + bv.y;
  o.z = (s.z - mean) * rstd * gv.z + bv.z;
  o.w = (s.w - mean) * rstd * gv.w + bv.w;
  ((float4*)(O + (size_t)wave * DMODEL))[lane] = o;
}

// ---------------------------------------------------------------------------
// degrees
// ---------------------------------------------------------------------------
__global__ void degree_kernel(const int* __restrict__ src,
                              const int* __restrict__ dst,
                              float* __restrict__ deg_out,
                              float* __restrict__ deg_in, int E) {
  int e = blockIdx.x * blockDim.x + threadIdx.x;
  if (e >= E) return;
  atomicAdd(deg_out + src[e], 1.0f);
  atomicAdd(deg_in + dst[e], 1.0f);
}

__global__ void deg_finalize_kernel(float* __restrict__ deg, int n) {
  int i = blockIdx.x * blockDim.x + threadIdx.x;
  if (i < n) deg[i] = rsqrtf(fmaxf(deg[i], 1.0f));
}

// ---------------------------------------------------------------------------
// GraphConv scatter: agg[dst] += h1[src] * degOutRs[src]  (4 floats/thread)
// ---------------------------------------------------------------------------
__global__ void gc_scatter_kernel(const float* __restrict__ h1,
                                  const int* __restrict__ src,
                                  const int* __restrict__ dst,
                                  const float* __restrict__ degOutRs,
                                  float* __restrict__ agg, int E) {
  int idx = blockIdx.x * blockDim.x + threadIdx.x;
  if (idx >= E * 32) return;
  int e = idx >> 5, q = idx & 31;
  int s = src[e], d = dst[e];
  float w = degOutRs[s];
  float4 hv = ((const float4*)(h1 + (size_t)s * DMODEL))[q];
  float* o = agg + (size_t)d * DMODEL + q * 4;
  atomicAdd(o + 0, hv.x * w);
  atomicAdd(o + 1, hv.y * w);
  atomicAdd(o + 2, hv.z * w);
  atomicAdd(o + 3, hv.w * w);
}

__global__ void scale_rows_kernel(float* __restrict__ A,
                                  const float* __restrict__ sc, int nrows) {
  int idx = blockIdx.x * blockDim.x + threadIdx.x;
  if (idx >= nrows * 32) return;
  int row = idx >> 5, q = idx & 31;
  float w = sc[row];
  float4* p = (float4*)(A + (size_t)row * DMODEL) + q;
  float4 v = *p;
  v.x *= w; v.y *= w; v.z *= w; v.w *= w;
  *p = v;
}

// ---------------------------------------------------------------------------
extern "C" void kernel_launch(void* const* d_in, const int* in_sizes, int n_in,
                              void* d_out, int out_size, void* d_ws, size_t ws_size,
                              hipStream_t stream) {
  const float* h   = (const float*)d_in[0];
  const int*   src = (const int*)d_in[1];
  const int*   dst = (const int*)d_in[2];
  const float* Wq = (const float*)d_in[3];  const float* bq = (const float*)d_in[4];
  const float* Wk = (const float*)d_in[5];  const float* bk = (const float*)d_in[6];
  const float* Wv = (const float*)d_in[7];  const float* bv = (const float*)d_in[8];
  const float* Wo = (const float*)d_in[9];  const float* bo = (const float*)d_in[10];
  const float* Wg = (const float*)d_in[11]; const float* bg = (const float*)d_in[12];
  const float* W1 = (const float*)d_in[13]; const float* b1 = (const float*)d_in[14];
  const float* W2 = (const float*)d_in[15]; const float* b2 = (const float*)d_in[16];
  const float* g1 = (const float*)d_in[17]; const float* be1 = (const float*)d_in[18];
  const float* g2 = (const float*)d_in[19]; const float* be2 = (const float*)d_in[20];
  const float* g3 = (const float*)d_in[21]; const float* be3 = (const float*)d_in[22];
  float* out = (float*)d_out;

  const int N = in_sizes[0] / DMODEL;   // 50000
  const int E = in_sizes[1];            // 600000
  const size_t ND = (size_t)N * DMODEL;
  const size_t NH = (size_t)N * NHEAD;
  const size_t EH = (size_t)E * NHEAD;

  float* ws       = (float*)d_ws;
  float* bufQ     = ws;
  float* bufK     = bufQ + ND;
  float* bufV     = bufK + ND;
  float* bufScore = bufV + ND;          // E*H
  float* bufSmax  = bufScore + EH;      // N*H
  float* bufDenom = bufSmax + NH;       // N*H
  float* bufAgg   = bufDenom + NH;      // N*D (attn agg, then graphconv agg)
  float* bufH1    = bufAgg + ND;
  float* bufH2    = bufH1 + ND;
  float* bufTmp   = bufH2 + ND;         // GEMM residual branch result
  float* bufDeg   = bufTmp + ND;        // [deg_out | deg_in], 2N
  float* bufMid   = bufDeg + 2 * (size_t)N; // N*512

  const int mtiles    = N / 16;               // 3125
  const int rowBlocks = (mtiles + 7) / 8;     // 8 row-tiles (128 rows) per block
  dim3 gblk(256);                             // 8 wave32s
  dim3 g_out128(rowBlocks, 128 / 16);
  dim3 g_out512(rowBlocks, 512 / 16);
  const size_t ldsK128 = (size_t)128 * 16 * sizeof(float);  //  8 KB W panel
  const size_t ldsK512 = (size_t)512 * 16 * sizeof(float);  // 32 KB W panel
  dim3 blk(256);
  const int gEH  = (int)((EH + 255) / 256);
  const int gE   = (E + 255) / 256;
  const int gE32 = (int)(((size_t)E * 32 + 255) / 256);
  const int gRow = (N + 7) / 8;               // 8 waves (rows) per block for LN

  // ---- Q, K, V projections (WMMA, async-LDS-staged W panels) ----
  gemm_bias_wmma<<<g_out128, gblk, ldsK128, stream>>>(h, Wq, bq, bufQ, N, DMODEL, DMODEL, 0);
  gemm_bias_wmma<<<g_out128, gblk, ldsK128, stream>>>(h, Wk, bk, bufK, N, DMODEL, DMODEL, 0);
  gemm_bias_wmma<<<g_out128, gblk, ldsK128, stream>>>(h, Wv, bv, bufV, N, DMODEL, DMODEL, 0);

  // ---- edge softmax over incoming edges of dst ----
  fill_f32<<<(int)((NH + 255) / 256), blk, 0, stream>>>(bufSmax, -3.0e38f, (long)NH);
  fill_f32<<<(int)((NH + 255) / 256), blk, 0, stream>>>(bufDenom, 0.0f, (long)NH);
  edge_score_kernel<<<gEH, blk, 0, stream>>>(bufQ, bufK, src, dst, bufScore, bufSmax, E);
  edge_exp_kernel<<<gEH, blk, 0, stream>>>(dst, bufScore, bufSmax, bufDenom, E);
  fill_f32<<<(int)((ND + 255) / 256), blk, 0, stream>>>(bufAgg, 0.0f, (long)ND);
  edge_scatter_kernel<<<gEH, blk, 0, stream>>>(bufV, src, dst, bufScore, bufDenom, bufAgg, E);

  // ---- output projection + residual LN -> h1 ----
  gemm_bias_wmma<<<g_out128, gblk, ldsK128, stream>>>(bufAgg, Wo, bo, bufTmp, N, DMODEL, DMODEL, 0);
  add_ln_kernel<<<gRow, blk, 0, stream>>>(h, bufTmp, g1, be1, bufH1, N);

  // ---- GraphConv (norm='both') ----
  fill_f32<<<(int)((2 * (size_t)N + 255) / 256), blk, 0, stream>>>(bufDeg, 0.0f, 2L * N);
  degree_kernel<<<gE, blk, 0, stream>>>(src, dst, bufDeg, bufDeg + N, E);
  deg_finalize_kernel<<<(2 * N + 255) / 256, blk, 0, stream>>>(bufDeg, 2 * N);
  fill_f32<<<(int)((ND + 255) / 256), blk, 0, stream>>>(bufAgg, 0.0f, (long)ND);
  gc_scatter_kernel<<<gE32, blk, 0, stream>>>(bufH1, src, dst, bufDeg, bufAgg, E);
  scale_rows_kernel<<<(int)(((size_t)N * 32 + 255) / 256), blk, 0, stream>>>(bufAgg, bufDeg + N, N);
  gemm_bias_wmma<<<g_out128, gblk, ldsK128, stream>>>(bufAgg, Wg, bg, bufTmp, N, DMODEL, DMODEL, 0);
  add_ln_kernel<<<gRow, blk, 0, stream>>>(bufH1, bufTmp, g2, be2, bufH2, N);

  // ---- FFN ----
  gemm_bias_wmma<<<g_out512, gblk, ldsK128, stream>>>(bufH2, W1, b1, bufMid, N, DMODEL, 512, 1);
  gemm_bias_wmma<<<g_out128, gblk, ldsK512, stream>>>(bufMid, W2, b2, bufTmp, N, 512, DMODEL, 0);
  add_ln_kernel<<<gRow, blk, 0, stream>>>(bufH2, bufTmp, g3, be3, out, N);
}